// QFTULayer_47055661695344
// MI455X (gfx1250) — compile-verified
//
#include <hip/hip_runtime.h>
#include <math.h>

typedef __attribute__((ext_vector_type(2))) float v2f;
typedef __attribute__((ext_vector_type(8))) float v8f;

#define RSQRT2 0.70710678118654752440f

struct c32 { float re, im; };

// Element [b][d] of _blk(e, c0, c3) = [[c0,0,0,0],[0,a,bb,0],[0,bb,a,0],[0,0,0,c3]]
// with a=(e+1)/2, bb=(e-1)/2.  All complex.
__device__ inline c32 blk_elem(float er, float ei, float c0r, float c0i,
                               float c3r, float c3i, int b, int d) {
    c32 v; v.re = 0.f; v.im = 0.f;
    if (b == 0 && d == 0) { v.re = c0r; v.im = c0i; }
    else if (b == 3 && d == 3) { v.re = c3r; v.im = c3i; }
    else if ((b == 1 || b == 2) && (d == 1 || d == 2)) {
        if (b == d) { v.re = (er + 1.f) * 0.5f; v.im = ei * 0.5f; }   // a
        else        { v.re = (er - 1.f) * 0.5f; v.im = ei * 0.5f; }   // bb
    }
    return v;
}

// M[r][k] for stage matrix.  Stages:
// 0:H0 1:U(t0) 2:U(t1) 3:U(t2) 4:H1 5:U(t3) 6:U(t4) 7:H2 8:U(t5) 9:H3 10:SW03 11:SW12
__device__ c32 stage_elem(int stage, int r, int k, const float* th) {
    c32 out; out.re = 0.f; out.im = 0.f;
    int hq = -1, sw = 0; float t = 0.f;
    switch (stage) {
        case 0: hq = 0; break;
        case 1: t = th[0]; break;
        case 2: t = th[1]; break;
        case 3: t = th[2]; break;
        case 4: hq = 1; break;
        case 5: t = th[3]; break;
        case 6: t = th[4]; break;
        case 7: hq = 2; break;
        case 8: t = th[5]; break;
        case 9: hq = 3; break;
        case 10: sw = 1; break;
        default: sw = 2; break;
    }
    if (hq >= 0) {                       // Hadamard on qubit hq (bit 3-hq)
        int bit = 3 - hq;
        int mask = 15 & ~(1 << bit);
        if ((r & mask) == (k & mask)) {
            float v = RSQRT2;
            if (((r >> bit) & 1) && ((k >> bit) & 1)) v = -v;
            out.re = v;
        }
        return out;
    }
    if (sw) {                            // swap permutation: (P@O)[r] = O[sw(r)]
        int b0 = (sw == 1) ? 3 : 2;
        int b1 = (sw == 1) ? 0 : 1;
        int d = (((r >> b0) ^ (r >> b1)) & 1);
        int rs = r ^ ((d << b0) | (d << b1));
        if (k == rs) out.re = 1.f;
        return out;
    }
    // U16(t) = kron(P00,U00) + kron(P11,U11) + kron(Ppp,Up) + kron(Pmm,I4)
    float sn, cs;
    sincosf(t, &sn, &cs);
    float epr = cs, epi = sn;            // e^{+it}
    float emr = cs, emi = -sn;           // e^{-it}
    int a = r >> 2, b = r & 3, c2 = k >> 2, d = k & 3;
    if (a == 0 && c2 == 0) {             // P00 * U00 = blk(em, 1, em)
        c32 u = blk_elem(emr, emi, 1.f, 0.f, emr, emi, b, d);
        out.re += u.re; out.im += u.im;
    }
    if (a == 3 && c2 == 3) {             // P11 * U11 = blk(ep, ep, 1)
        c32 u = blk_elem(epr, epi, epr, epi, 1.f, 0.f, b, d);
        out.re += u.re; out.im += u.im;
    }
    float spa = (a == 1 || a == 2) ? RSQRT2 : 0.f;
    float spc = (c2 == 1 || c2 == 2) ? RSQRT2 : 0.f;
    float sma = (a == 1) ? RSQRT2 : ((a == 2) ? -RSQRT2 : 0.f);
    float smc = (c2 == 1) ? RSQRT2 : ((c2 == 2) ? -RSQRT2 : 0.f);
    if (b == d) {
        float ppp = spa * spc;           // Ppp * Up, Up = diag(ep,1,1,em)
        float upr = 1.f, upi = 0.f;
        if (b == 0) { upr = epr; upi = epi; }
        else if (b == 3) { upr = emr; upi = emi; }
        out.re += ppp * upr; out.im += ppp * upi;
        out.re += sma * smc;             // Pmm * I4
    }
    return out;
}

__global__ void build_gate(const float* __restrict__ thetas, float* __restrict__ gws) {
    __shared__ float cre[256], cim[256];
    __shared__ float th[6];
    int tid = threadIdx.x;
    if (tid < 6) th[tid] = thetas[tid];
    int r = tid >> 4, c = tid & 15;
    cre[tid] = (r == c) ? 1.f : 0.f;
    cim[tid] = 0.f;
    __syncthreads();
    for (int stage = 0; stage < 12; ++stage) {
        float ar = 0.f, ai = 0.f;
        for (int k = 0; k < 16; ++k) {
            c32 m = stage_elem(stage, r, k, th);
            float br = cre[k * 16 + c], bi = cim[k * 16 + c];
            ar += m.re * br - m.im * bi;
            ai += m.re * bi + m.im * br;
        }
        __syncthreads();
        cre[tid] = ar; cim[tid] = ai;
        __syncthreads();
    }
    gws[tid]       = cre[tid];   // Gre, row-major 16x16
    gws[256 + tid] = cim[tid];   // Gim
}

// One wave per batch b; 8 waves per block. OUT[16][128] = G(16x16 cplx) x X(16x128 cplx).
__global__ __launch_bounds__(256) void apply_gate(
    const float* __restrict__ xre, const float* __restrict__ xim,
    const float* __restrict__ gws, float* __restrict__ out)
{
    const int lane  = threadIdx.x & 31;
    const int b     = blockIdx.x * 8 + (threadIdx.x >> 5);
    const int nrow  = lane & 15;     // A operand: M index; B operand: N index; D: N index
    const int khalf = lane >> 4;     // selects K pair within each K=4 chunk

    // A-operand layout (16x4 f32): lane = m + 16*(k>>1), reg = k&1.
    // For K-chunk q (cols 4q..4q+3 of G): this lane holds G[nrow][4q+2*khalf + {0,1}].
    v2f agr[4], agi[4], agin[4];
#pragma unroll
    for (int q = 0; q < 4; ++q) {
        int k0 = 4 * q + 2 * khalf;
        agr[q].x  = gws[nrow * 16 + k0];
        agr[q].y  = gws[nrow * 16 + k0 + 1];
        agi[q].x  = gws[256 + nrow * 16 + k0];
        agi[q].y  = gws[256 + nrow * 16 + k0 + 1];
        agin[q].x = -agi[q].x;          // f32 WMMA has no A-negate (NEG=CNeg only)
        agin[q].y = -agi[q].y;
    }

    const long base = (long)b * (16 * 128);
#pragma unroll 1
    for (int tile = 0; tile < 8; ++tile) {
        const int c0 = tile * 16;
        v8f dre = {}; v8f dim = {};
#pragma unroll
        for (int q = 0; q < 4; ++q) {
            int k0 = 4 * q + 2 * khalf;
            // B-operand layout (4x16 f32): lane = n + 16*(k>>1), reg = k&1.
            const float* pr = xre + base + (long)k0 * 128 + c0 + nrow;
            const float* pi = xim + base + (long)k0 * 128 + c0 + nrow;
            v2f bre, bim;
            bre.x = pr[0]; bre.y = pr[128];
            bim.x = pi[0]; bim.y = pi[128];
            // dre += Gre*Xre - Gim*Xim ; dim += Gre*Xim + Gim*Xre
            dre = __builtin_amdgcn_wmma_f32_16x16x4_f32(false, agr[q],  false, bre,
                                                        (short)0, dre, false, false);
            dre = __builtin_amdgcn_wmma_f32_16x16x4_f32(false, agin[q], false, bim,
                                                        (short)0, dre, false, false);
            dim = __builtin_amdgcn_wmma_f32_16x16x4_f32(false, agr[q],  false, bim,
                                                        (short)0, dim, false, false);
            dim = __builtin_amdgcn_wmma_f32_16x16x4_f32(false, agi[q],  false, bre,
                                                        (short)0, dim, false, false);
        }
        // D layout: reg r, lane l -> m = r + 8*(l>>4), n = l&15. Store interleaved complex.
#pragma unroll
        for (int r = 0; r < 8; ++r) {
            int m = r + 8 * khalf;
            long idx = base + (long)m * 128 + c0 + nrow;   // complex element index
            float2 v; v.x = dre[r]; v.y = dim[r];
            *reinterpret_cast<float2*>(out + 2 * idx) = v;
        }
    }
}

extern "C" void kernel_launch(void* const* d_in, const int* in_sizes, int n_in,
                              void* d_out, int out_size, void* d_ws, size_t ws_size,
                              hipStream_t stream) {
    const float* xre    = (const float*)d_in[0];   // [8192,16,128] f32
    const float* xim    = (const float*)d_in[1];   // [8192,16,128] f32
    const float* thetas = (const float*)d_in[2];   // [6] f32
    float* outp = (float*)d_out;                   // complex64 -> interleaved f32 pairs
    float* gws  = (float*)d_ws;                    // 512 floats: Gre | Gim

    build_gate<<<1, 256, 0, stream>>>(thetas, gws);
    apply_gate<<<1024, 256, 0, stream>>>(xre, xim, gws, outp);
}